// GravNet_20117626814684
// MI455X (gfx1250) — compile-verified
//
#include <hip/hip_runtime.h>
#include <hip/hip_bf16.h>

typedef __attribute__((ext_vector_type(2))) float v2f;
typedef __attribute__((ext_vector_type(8))) float v8f;

#define BATCH 64
#define NNODE 1024
#define KNN   20
#define NEG_SLOPE 0.01f
#define NTOT (BATCH * NNODE)
#define KCH  128   // K-chunk staged in LDS per iteration (floats)

// ---------------------------------------------------------------------------
// s/h projection: per node, 5 dot products over cin (3 spatial + 2 feature).
// sh layout: [node][8] -> {s0,s1,s2,h0,h1,pad,pad,pad}
// ---------------------------------------------------------------------------
__global__ void sh_kernel(const float* __restrict__ x, int cin,
                          const float* __restrict__ ws, const float* __restrict__ bs,
                          const float* __restrict__ wh, const float* __restrict__ bh,
                          float* __restrict__ sh) {
  int node = blockIdx.x * blockDim.x + threadIdx.x;
  if (node >= NTOT) return;
  const float* xr = x + (size_t)node * cin;
  float a0 = bs[0], a1 = bs[1], a2 = bs[2], a3 = bh[0], a4 = bh[1];
  for (int c = 0; c < cin; ++c) {
    float xv = xr[c];
    a0 += xv * ws[c * 3 + 0];
    a1 += xv * ws[c * 3 + 1];
    a2 += xv * ws[c * 3 + 2];
    a3 += xv * wh[c * 2 + 0];
    a4 += xv * wh[c * 2 + 1];
  }
  float* o = sh + (size_t)node * 8;
  o[0] = a0; o[1] = a1; o[2] = a2; o[3] = a3; o[4] = a4;
}

// ---------------------------------------------------------------------------
// Top-K (K=20) nearest neighbors in 3-D s-space, per node, within batch.
// Block = 256 threads handles 256 nodes of one batch; batch's 1024 s-vectors
// staged in LDS (12 KB of the 320 KB/WGP). Register-resident unrolled
// insertion sort (stable: strict '<' matches jax.lax.top_k tie-break).
// ---------------------------------------------------------------------------
__global__ void topk_kernel(const float* __restrict__ sh,
                            float* __restrict__ knn_d2, int* __restrict__ knn_idx) {
  __shared__ float s0[NNODE], s1[NNODE], s2[NNODE];
  int b = blockIdx.x >> 2;
  int chunk = blockIdx.x & 3;
  int t = threadIdx.x;
  for (int i = t; i < NNODE; i += 256) {
    const float* p = sh + ((size_t)(b * NNODE + i)) * 8;
    s0[i] = p[0]; s1[i] = p[1]; s2[i] = p[2];
  }
  __syncthreads();
  int n = chunk * 256 + t;
  float q0 = s0[n], q1 = s1[n], q2 = s2[n];
  float bd[KNN]; int bi[KNN];
#pragma unroll
  for (int j = 0; j < KNN; ++j) { bd[j] = 3.4e38f; bi[j] = 0; }
  for (int m = 0; m < NNODE; ++m) {
    float d0 = q0 - s0[m], d1 = q1 - s1[m], d2v = q2 - s2[m];
    float d = d0 * d0 + d1 * d1 + d2v * d2v;
    if (d < bd[KNN - 1]) {
      bd[KNN - 1] = d; bi[KNN - 1] = m;
#pragma unroll
      for (int j = KNN - 1; j > 0; --j) {
        if (bd[j] < bd[j - 1]) {
          float td = bd[j]; bd[j] = bd[j - 1]; bd[j - 1] = td;
          int ti = bi[j]; bi[j] = bi[j - 1]; bi[j - 1] = ti;
        }
      }
    }
  }
  size_t base = (size_t)(b * NNODE + n) * KNN;
#pragma unroll
  for (int j = 0; j < KNN; ++j) { knn_d2[base + j] = bd[j]; knn_idx[base + j] = bi[j]; }
}

// ---------------------------------------------------------------------------
// Aggregate messages: w = exp(-10*max(d2,0)); mean & max over K of h_j*w.
// Writes xcat row = [mean0, mean1, max0, max1, x[0..cin-1]]  (kdim = cin+4).
// ---------------------------------------------------------------------------
__global__ void agg_kernel(const float* __restrict__ x, int cin,
                           const float* __restrict__ sh,
                           const float* __restrict__ knn_d2,
                           const int* __restrict__ knn_idx,
                           float* __restrict__ xcat, int kdim) {
  int node = blockIdx.x * blockDim.x + threadIdx.x;
  if (node >= NTOT) return;
  int b = node >> 10;
  size_t base = (size_t)node * KNN;
  float mean0 = 0.f, mean1 = 0.f, mx0 = -3.4e38f, mx1 = -3.4e38f;
  for (int k = 0; k < KNN; ++k) {
    int mi = knn_idx[base + k];
    float d2 = knn_d2[base + k];
    float w = expf(-10.f * fmaxf(d2, 0.f));
    const float* hp = sh + ((size_t)((b << 10) + mi)) * 8 + 3;
    float v0 = hp[0] * w, v1 = hp[1] * w;
    mean0 += v0; mean1 += v1;
    mx0 = fmaxf(mx0, v0); mx1 = fmaxf(mx1, v1);
  }
  float* o = xcat + (size_t)node * kdim;
  o[0] = mean0 * (1.f / KNN);
  o[1] = mean1 * (1.f / KNN);
  o[2] = mx0;
  o[3] = mx1;
  const float* xr = x + (size_t)node * cin;
  for (int c = 0; c < cin; ++c) o[4 + c] = xr[c];
}

// ---------------------------------------------------------------------------
// Dense transform on the fp32 WMMA pipe: Y = A(kdim) @ W(kdim x ncol) + b,
// optional leaky-ReLU.  V_WMMA_F32_16X16X4_F32.
//
// Block = 8 waves = 8 row-tiles x ONE 16*NT-column group; the W panel for
// the group is staged in LDS in K-chunks of KCH (32 KB for NT=4), stored as
// interleaved (k,k+1) pairs so a lane's B fragment is one aligned
// ds_load_b64. Column guard is resolved at fill time (zeros), so the inner
// loop is: 1 global_load_b64 (A) + NT ds_load_b64 (B) + NT independent WMMAs.
//   A lane layout: lane m=lane&15 holds A[m][kb],A[m][kb+1], kb=(lane>>4)*2
//   B lane layout: lane n=lane&15 holds B[kb][n],B[kb+1][n]
//   D: VGPR i -> row i (lanes 0-15) / row i+8 (lanes 16-31), col = lane&15
// ---------------------------------------------------------------------------
template <int NT>
__global__ void gemm_wmma_kernel(const float* __restrict__ A, int kdim,
                                 const float* __restrict__ W,
                                 const float* __restrict__ bias,
                                 float* __restrict__ Y, int ncol, int groups_n,
                                 int lrelu) {
  const int W16 = 16 * NT;                    // panel width in columns
  __shared__ v2f wpair[(KCH / 2) * 16 * NT];  // 32 KB for NT=4, 8 KB for NT=1

  int rg = blockIdx.x / groups_n;             // row group (8 row-tiles)
  int tg = blockIdx.x - rg * groups_n;        // column group
  int col0 = tg * W16;

  int t = threadIdx.x;
  int wv = t >> 5;
  int lane = t & 31;
  int mlane = lane & 15;
  int kb = (lane >> 4) << 1;                  // 0 or 2
  int row0 = (rg * 8 + wv) << 4;

  v8f acc[NT];
#pragma unroll
  for (int nt = 0; nt < NT; ++nt) acc[nt] = (v8f){};

  const float* arow = A + (size_t)(row0 + mlane) * kdim;

  for (int kc = 0; kc < kdim; kc += KCH) {
    int klen = (kdim - kc) < KCH ? (kdim - kc) : KCH;  // always even (20/260)

    // --- cooperative W panel fill: pair-interleaved, guarded with zeros ---
    int npair = (klen >> 1) * W16;
    for (int idx = t; idx < npair; idx += 256) {
      int kp = idx / W16;
      int c = idx - kp * W16;
      int col = col0 + c;
      v2f wv2;
      if (col < ncol) {
        wv2.x = W[(size_t)(kc + 2 * kp) * ncol + col];
        wv2.y = W[(size_t)(kc + 2 * kp + 1) * ncol + col];
      } else {
        wv2.x = 0.f; wv2.y = 0.f;
      }
      wpair[kp * W16 + c] = wv2;
    }
    __syncthreads();

    // --- WMMA over this K chunk ---
    for (int kt = 0; kt < klen; kt += 4) {
      v2f av = *(const v2f*)(arow + kc + kt + kb);
      if (kc + kt + 64 < kdim) __builtin_prefetch(arow + kc + kt + 64, 0, 1);
      v2f bv[NT];
#pragma unroll
      for (int nt = 0; nt < NT; ++nt)
        bv[nt] = wpair[((kt + kb) >> 1) * W16 + nt * 16 + mlane];
#pragma unroll
      for (int nt = 0; nt < NT; ++nt)
        acc[nt] = __builtin_amdgcn_wmma_f32_16x16x4_f32(
            /*neg_a=*/false, av, /*neg_b=*/false, bv[nt],
            /*c_mod=*/(short)0, acc[nt], /*reuse_a=*/false, /*reuse_b=*/false);
    }
    __syncthreads();
  }

  int rbase = row0 + ((lane >> 4) << 3);  // rows i / i+8 split across lane halves
#pragma unroll
  for (int nt = 0; nt < NT; ++nt) {
    int col = col0 + nt * 16 + mlane;
    if (col < ncol) {
      float bb = bias[col];
#pragma unroll
      for (int i = 0; i < 8; ++i) {
        float v = acc[nt][i] + bb;
        if (lrelu) v = (v > 0.f) ? v : NEG_SLOPE * v;
        Y[(size_t)(rbase + i) * ncol + col] = v;
      }
    }
  }
}

// ---------------------------------------------------------------------------
// Final gather: output = layer2_out[:, node 0 of each batch, :5]
// ---------------------------------------------------------------------------
__global__ void gather_out_kernel(const float* __restrict__ y, float* __restrict__ out) {
  int i = blockIdx.x * blockDim.x + threadIdx.x;
  if (i >= BATCH * 5) return;
  int b = i / 5, j = i - b * 5;
  out[i] = y[(size_t)(b * NNODE) * 5 + j];
}

// ---------------------------------------------------------------------------
extern "C" void kernel_launch(void* const* d_in, const int* in_sizes, int n_in,
                              void* d_out, int out_size, void* d_ws, size_t ws_size,
                              hipStream_t stream) {
  const float* x = (const float*)d_in[0];
  // d_in: 0=x, 1=batch, then per layer l: ws,bs,wh,bh,wl,bl at 2+6l..7+6l
  const float* ws[3], *bs[3], *wh[3], *bh[3], *wl[3], *bl[3];
  for (int l = 0; l < 3; ++l) {
    ws[l] = (const float*)d_in[2 + 6 * l];
    bs[l] = (const float*)d_in[3 + 6 * l];
    wh[l] = (const float*)d_in[4 + 6 * l];
    bh[l] = (const float*)d_in[5 + 6 * l];
    wl[l] = (const float*)d_in[6 + 6 * l];
    bl[l] = (const float*)d_in[7 + 6 * l];
  }

  // Workspace carve-up (floats): ~148 MB total
  float* p = (float*)d_ws;
  float* sh      = p; p += (size_t)NTOT * 8;          //  2.1 MB
  float* knn_d2  = p; p += (size_t)NTOT * KNN;        //  5.2 MB
  int*   knn_idx = (int*)p; p += (size_t)NTOT * KNN;  //  5.2 MB
  float* xcat    = p; p += (size_t)NTOT * 260;        // 68.2 MB
  float* y       = p; p += (size_t)NTOT * 256;        // 67.1 MB

  const int cins[3]  = {16, 256, 256};
  const int ncols[3] = {256, 256, 5};

  for (int l = 0; l < 3; ++l) {
    const float* xin = (l == 0) ? x : y;
    int cin = cins[l], kdim = cin + 4, ncol = ncols[l];

    sh_kernel<<<NTOT / 256, 256, 0, stream>>>(xin, cin, ws[l], bs[l], wh[l], bh[l], sh);
    topk_kernel<<<BATCH * 4, 256, 0, stream>>>(sh, knn_d2, knn_idx);
    agg_kernel<<<NTOT / 256, 256, 0, stream>>>(xin, cin, sh, knn_d2, knn_idx, xcat, kdim);

    int row_groups = (NTOT / 16) / 8;  // 512 blocks of 8 row-tiles
    if (ncol == 256) {
      int groups_n = 256 / 64;  // 4 column groups of 64
      gemm_wmma_kernel<4><<<row_groups * groups_n, 256, 0, stream>>>(
          xcat, kdim, wl[l], bl[l], y, ncol, groups_n, (l < 2) ? 1 : 0);
    } else {
      // ncol = 5: one 16-wide column group with guard
      gemm_wmma_kernel<1><<<row_groups, 256, 0, stream>>>(
          xcat, kdim, wl[l], bl[l], y, ncol, 1, 0);
    }
  }
  gather_out_kernel<<<2, 256, 0, stream>>>(y, (float*)d_out);
}